// DotProductPredictor_34634616275547
// MI455X (gfx1250) — compile-verified
//
#include <hip/hip_runtime.h>

typedef __attribute__((ext_vector_type(2))) float v2f;
typedef __attribute__((ext_vector_type(8))) float v8f;

#define D_FEAT 32
#define EDGES_PER_WAVE 16

// One wave computes 16 edge dot-products as a 16x16 GEMM diagonal:
//   A (16x32) = h[src[e0..e0+15]], B (32x16) = h[dst[e0..e0+15]]^T
//   D = sum_k A(16x4) x B(4x16)  via 8x V_WMMA_F32_16X16X4_F32 (exact f32 FMA, RNE)
//   score[e0+m] = D[m][m]
__global__ __launch_bounds__(256) void edge_dot_wmma_kernel(
    const float* __restrict__ h,
    const long long* __restrict__ src,
    const long long* __restrict__ dst,
    float* __restrict__ out,
    int n_edges)
{
    const int lane = (int)(threadIdx.x & 31u);
    const int wave = (int)((blockIdx.x * blockDim.x + threadIdx.x) >> 5);
    const int e0   = wave * EDGES_PER_WAVE;
    if (e0 >= n_edges) return;          // wave-uniform exit: EXEC stays all-1s below

    const int m    = lane & 15;         // edge slot within the 16-edge tile (M for A, N for B)
    const int half = lane >> 4;         // 0 -> K pair {4k,4k+1}; 1 -> K pair {4k+2,4k+3}

    int e = e0 + m;
    if (e > n_edges - 1) e = n_edges - 1;   // clamp tail so all lanes load valid rows

    const long long srow = src[e];
    const long long drow = dst[e];
    const float* __restrict__ pa = h + srow * D_FEAT + 2 * half;
    const float* __restrict__ pb = h + drow * D_FEAT + 2 * half;

    // Each lane gathers the 16 features it feeds into the 8 K-steps:
    // stride-4 float2 pairs -> 8B-aligned global_load_b64, all L2-resident (h = 12.8MB < 192MB L2).
    v2f a[8], b[8];
#pragma unroll
    for (int k = 0; k < 8; ++k) {
        a[k] = *(const v2f*)(pa + 4 * k);
        b[k] = *(const v2f*)(pb + 4 * k);
    }

    // Accumulate D = sum over 8 K-chunks of A(16x4) x B(4x16) in f32.
    v8f c = {};
#pragma unroll
    for (int k = 0; k < 8; ++k) {
        c = __builtin_amdgcn_wmma_f32_16x16x4_f32(
                /*neg_a=*/false, a[k],
                /*neg_b=*/false, b[k],
                /*c_mod=*/(short)0, c,
                /*reuse_a=*/false, /*reuse_b=*/false);
    }

    // Diagonal extraction from the 16x16 f32 C/D layout:
    //   vgpr r, lanes 0-15  -> M=r,   N=lane      => diag m=r  at lane r
    //   vgpr r, lanes 16-31 -> M=r+8, N=lane-16   => diag m=r+8 at lane r+24
    float val = 0.0f;
    int eo = -1;
#pragma unroll
    for (int r = 0; r < 8; ++r) {
        if (lane == r)      { val = c[r]; eo = r; }
        if (lane == r + 24) { val = c[r]; eo = r + 8; }
    }
    if (eo >= 0) {
        const int eidx = e0 + eo;
        if (eidx < n_edges) out[eidx] = val;
    }
}

extern "C" void kernel_launch(void* const* d_in, const int* in_sizes, int n_in,
                              void* d_out, int out_size, void* d_ws, size_t ws_size,
                              hipStream_t stream) {
    (void)n_in; (void)out_size; (void)d_ws; (void)ws_size;

    const float*     h   = (const float*)d_in[0];       // [N_NODES, 32] f32
    const long long* src = (const long long*)d_in[1];   // [E] i64
    const long long* dst = (const long long*)d_in[2];   // [E] i64
    float*           out = (float*)d_out;               // [E] f32

    const int n_edges = in_sizes[1];
    const int waves   = (n_edges + EDGES_PER_WAVE - 1) / EDGES_PER_WAVE;
    const int blocks  = (waves + 7) / 8;                // 8 wave32s per 256-thread block

    edge_dot_wmma_kernel<<<blocks, 256, 0, stream>>>(h, src, dst, out, n_edges);
}